// CustomInputLayer_8358006358532
// MI455X (gfx1250) — compile-verified
//
#include <hip/hip_runtime.h>
#include <hip/hip_bf16.h>
#include <stdint.h>

// Problem constants (from the reference): N=1M points, C=64, GRID=128, BATCH=4
#define C_FEAT     64
#define GRIDD      128
#define VOL        (GRIDD * GRIDD * GRIDD)          // 2,097,152
#define NBATCH     4
#define TOTAL_VOX  ((uint32_t)NBATCH * (uint32_t)VOL) // 8,388,608 (< 2^23 -> exact in f32)
#define WORDS      (TOTAL_VOX / 32u)                // 262,144 bitmask words
#define SCAN_BLOCK 256
#define SCAN_GRID  (WORDS / SCAN_BLOCK)             // 1024

typedef float    f4  __attribute__((ext_vector_type(4)));
typedef uint32_t u4v __attribute__((ext_vector_type(4)));
typedef int      i4v __attribute__((ext_vector_type(4)));

// ---------------- CDNA5 async global->LDS copy (B128) -----------------------
__device__ __forceinline__ void async_copy16(const float* g, float* l) {
#if __has_builtin(__builtin_amdgcn_global_load_async_to_lds_b128)
  // clang signature: (v4i addrspace(1)*, v4i addrspace(3)*, Imm offset, Imm cpol)
  float* gnc = const_cast<float*>(g);
  __builtin_amdgcn_global_load_async_to_lds_b128(
      (__attribute__((address_space(1))) i4v*)gnc,
      (__attribute__((address_space(3))) i4v*)l, 0, 0);
#else
  // VDST = per-lane LDS byte address, VADDR = 64-bit global address
  unsigned lds_off = (unsigned)(uintptr_t)l;
  asm volatile("global_load_async_to_lds_b128 %0, %1, off"
               :: "v"(lds_off), "v"(g) : "memory");
#endif
}

#if __has_builtin(__builtin_amdgcn_s_wait_asynccnt)
#define WAIT_ASYNC(n) __builtin_amdgcn_s_wait_asynccnt(n)
#else
#define WAIT_ASYNC(n) asm volatile("s_wait_asynccnt %0" :: "i"(n) : "memory")
#endif

__device__ __forceinline__ void atomic_add_f32(float* p, float v) {
#if __has_builtin(__builtin_amdgcn_global_atomic_fadd_f32)
  (void)__builtin_amdgcn_global_atomic_fadd_f32(
      (__attribute__((address_space(1))) float*)p, v);
#else
  atomicAdd(p, v);  // lowers to global_atomic_add_f32 (no return) on gfx1250
#endif
}

// ---------------- wave32 inclusive scan -------------------------------------
__device__ __forceinline__ uint32_t wave_incl_scan(uint32_t x) {
  const int lane = threadIdx.x & 31;
#pragma unroll
  for (int off = 1; off < 32; off <<= 1) {
    uint32_t y = __shfl_up(x, off, 32);
    if (lane >= off) x += y;
  }
  return x;
}

// ---------------- kernel 0: init output (-1 ids, 0 features) + zero mask ----
__global__ void __launch_bounds__(256)
k_init(float* __restrict__ out, uint32_t n_pts, u4v* __restrict__ mask4) {
  const size_t stride = (size_t)gridDim.x * blockDim.x;
  const size_t tid    = (size_t)blockIdx.x * blockDim.x + threadIdx.x;
  const size_t ids4   = n_pts / 4;                          // ids region in float4s
  const size_t tot4   = (size_t)n_pts * (C_FEAT + 1) / 4;   // whole output in float4s
  f4* out4 = (f4*)out;
  const f4 neg1 = {-1.f, -1.f, -1.f, -1.f};
  const f4 zero = {0.f, 0.f, 0.f, 0.f};
  for (size_t i = tid; i < tot4; i += stride)
    __builtin_nontemporal_store(i < ids4 ? neg1 : zero, &out4[i]);
  const u4v z = {0u, 0u, 0u, 0u};
  for (size_t i = tid; i < (WORDS / 4); i += stride)
    __builtin_nontemporal_store(z, &mask4[i]);
}

// ---------------- kernel 1: linear ids + occupancy bitmask ------------------
__global__ void __launch_bounds__(256)
k_mark(const int4* __restrict__ coords, uint32_t n,
       uint32_t* __restrict__ mask, uint32_t* __restrict__ lin) {
  const uint32_t i = blockIdx.x * blockDim.x + threadIdx.x;
  if (i >= n) return;
  int4 c = coords[i];  // [x, y, z, batch]
  uint32_t l = (uint32_t)c.w * (uint32_t)VOL +
               (uint32_t)c.x * (uint32_t)(GRIDD * GRIDD) +
               (uint32_t)c.y * (uint32_t)GRIDD + (uint32_t)c.z;
  lin[i] = l;
  atomicOr(&mask[l >> 5], 1u << (l & 31));
}

// ---------------- kernel 2: per-block exclusive scan of word popcounts ------
__global__ void __launch_bounds__(256)
k_scan(const uint32_t* __restrict__ mask, uint32_t* __restrict__ wordPrefix,
       uint32_t* __restrict__ blockSums) {
  const uint32_t gid = blockIdx.x * (uint32_t)SCAN_BLOCK + threadIdx.x;
  const int lane = threadIdx.x & 31, wid = threadIdx.x >> 5;
  uint32_t cnt  = (uint32_t)__popc(mask[gid]);
  uint32_t incl = wave_incl_scan(cnt);
  __shared__ uint32_t wtot[8];
  if (lane == 31) wtot[wid] = incl;
  __syncthreads();
  if (wid == 0) {
    uint32_t v = (lane < 8) ? wtot[lane] : 0u;
    uint32_t s = wave_incl_scan(v);
    if (lane < 8) wtot[lane] = s - v;   // exclusive wave offsets
  }
  __syncthreads();
  uint32_t excl = incl - cnt + wtot[wid];
  wordPrefix[gid] = excl;               // block-local exclusive prefix
  if (threadIdx.x == SCAN_BLOCK - 1) blockSums[blockIdx.x] = excl + cnt;
}

// ---------------- kernel 3: scan the 1024 block sums (single block) ---------
__global__ void __launch_bounds__(1024)
k_scan_sums(const uint32_t* __restrict__ blockSums, uint32_t* __restrict__ blockOff) {
  const int lane = threadIdx.x & 31, wid = threadIdx.x >> 5;
  uint32_t v    = blockSums[threadIdx.x];
  uint32_t incl = wave_incl_scan(v);
  __shared__ uint32_t wtot[32];
  if (lane == 31) wtot[wid] = incl;
  __syncthreads();
  if (wid == 0) {
    uint32_t t = wtot[lane];
    uint32_t s = wave_incl_scan(t);
    wtot[lane] = s - t;
  }
  __syncthreads();
  blockOff[threadIdx.x] = incl - v + wtot[wid];   // global exclusive
}

// ---------------- kernel 4: globalize prefix + emit sorted unique ids -------
__global__ void __launch_bounds__(256)
k_finalize(const uint32_t* __restrict__ mask, uint32_t* __restrict__ wordPrefix,
           const uint32_t* __restrict__ blockOff, float* __restrict__ outIds) {
  const uint32_t gid = blockIdx.x * (uint32_t)SCAN_BLOCK + threadIdx.x;
  uint32_t wp = wordPrefix[gid] + blockOff[gid >> 8];
  wordPrefix[gid] = wp;                 // now: #set bits in words [0, gid)
  uint32_t m = mask[gid];
  const uint32_t base = gid << 5;
  while (m) {                           // sorted ascending; ids < 2^23 exact in f32
    uint32_t b = (uint32_t)__builtin_ctz(m);
    m &= m - 1u;
    outIds[wp++] = (float)(base + b);
  }
}

// ---------------- kernel 5: scatter-sum features (async-LDS pipelined) ------
#define WPB 8  // waves per 256-thread block
__global__ void __launch_bounds__(256)
k_scatter(const float* __restrict__ feat, const uint32_t* __restrict__ lin,
          const uint32_t* __restrict__ mask, const uint32_t* __restrict__ wordPrefix,
          float* __restrict__ outFeat, uint32_t nPairs) {
  __shared__ float buf[WPB][2][128];    // 2 x 512B double buffer per wave
  const int lane = threadIdx.x & 31;
  const int wid  = threadIdx.x >> 5;
  uint32_t wave   = blockIdx.x * (blockDim.x >> 5) + wid;
  const uint32_t wstride = gridDim.x * (blockDim.x >> 5);

  uint32_t pair = wave;
  if (pair < nPairs)  // prologue: stage first pair (2 points = 512B, 16B/lane)
    async_copy16(feat + (size_t)pair * (2 * C_FEAT) + lane * 4, &buf[wid][0][lane * 4]);

  int sel = 0;
  for (; pair < nPairs; pair += wstride) {
    uint32_t nxt = pair + wstride;
    if (nxt < nPairs) {
      async_copy16(feat + (size_t)nxt * (2 * C_FEAT) + lane * 4, &buf[wid][sel ^ 1][lane * 4]);
      WAIT_ASYNC(1);   // async loads complete in order: current buffer is ready
    } else {
      WAIT_ASYNC(0);
    }
    // rank lookup for this lane's point (lanes 0-15 -> point0, 16-31 -> point1)
    uint32_t p   = pair * 2u + (uint32_t)(lane >> 4);
    uint32_t l   = lin[p];
    uint32_t w   = l >> 5, off = l & 31u;
    uint32_t inv = wordPrefix[w] + (uint32_t)__popc(mask[w] & ((1u << off) - 1u));
    float*       dst = outFeat + (size_t)inv * C_FEAT + (size_t)(lane & 15) * 4;
    const float* src = &buf[wid][sel][lane * 4];
#pragma unroll
    for (int j = 0; j < 4; ++j) atomic_add_f32(dst + j, src[j]);
    sel ^= 1;
  }
}

// ---------------- host-side launcher ----------------------------------------
extern "C" void kernel_launch(void* const* d_in, const int* in_sizes, int n_in,
                              void* d_out, int out_size, void* d_ws, size_t ws_size,
                              hipStream_t stream) {
  const int*   coords = (const int*)d_in[0];    // [N,4] as int32
  const float* feat   = (const float*)d_in[1];  // [N,64]
  const uint32_t n    = (uint32_t)(in_sizes[0] / 4);
  float* out = (float*)d_out;                   // [N ids | N*64 features]

  uint8_t* ws = (uint8_t*)d_ws;
  uint32_t* mask       = (uint32_t*)ws;                          // 1 MB
  uint32_t* wordPrefix = (uint32_t*)(ws + (size_t)WORDS * 4);    // 1 MB
  uint32_t* blockSums  = (uint32_t*)(ws + (size_t)WORDS * 8);    // 4 KB
  uint32_t* blockOff   = blockSums + SCAN_GRID;                  // 4 KB
  uint32_t* lin        = blockOff + SCAN_GRID;                   // 4 MB

  k_init<<<2048, 256, 0, stream>>>(out, n, (u4v*)mask);
  k_mark<<<(n + 255) / 256, 256, 0, stream>>>((const int4*)coords, n, mask, lin);
  k_scan<<<SCAN_GRID, SCAN_BLOCK, 0, stream>>>(mask, wordPrefix, blockSums);
  k_scan_sums<<<1, 1024, 0, stream>>>(blockSums, blockOff);
  k_finalize<<<SCAN_GRID, SCAN_BLOCK, 0, stream>>>(mask, wordPrefix, blockOff, out);
  k_scatter<<<4096, 256, 0, stream>>>(feat, lin, mask, wordPrefix, out + n, n / 2);
}